// OmniVoiceKvWrapper_44925357916268
// MI455X (gfx1250) — compile-verified
//
#include <hip/hip_runtime.h>
#include <stdint.h>

// Problem constants from the reference:
//   L=8, B=2, H=8, S_FULL=4096, S_NEW=4, D=128
// Output: stack((present_k, present_v)) -> (2, L, B, H, S_FULL, D) fp32.
// Pure bandwidth problem: ~1.07 GB traffic, floor ~46 us @ 23.3 TB/s.
// Zero FLOPs -> WMMA inapplicable; the CDNA5 path that matters is the
// async-tensor DMA pipe (global<->LDS, ASYNCcnt) with NT cache hints.

constexpr int kL = 8;
constexpr int kB = 2;
constexpr int kH = 8;
constexpr int kSF = 4096;
constexpr int kSN = 4;
constexpr int kD = 128;
constexpr int kD4 = kD / 4;                            // float4 per row = 32
constexpr long kN4 = (long)kL * kB * kH * kSF * kD4;   // float4 per tensor = 16,777,216

constexpr int BULK_THREADS = 256;   // 8 wave32 per block
constexpr int BATCH = 4;            // async chunks in flight per thread per direction
constexpr int BULK_BLOCKS = 4096;   // x-dim; grid.y=2 picks K/V -> 8192 WGs total

// ---------------------------------------------------------------------------
// Kernel 1: bulk copy past_k -> out[0], past_v -> out[1] via the gfx1250
// async DMA path, double-buffered so each s_wait_asynccnt 0 window carries
// BATCH 16B loads AND BATCH 16B stores per lane concurrently.
// Per-thread private LDS slots -> no barriers needed.
// ---------------------------------------------------------------------------
__global__ void kv_bulk_copy_async(const float4* __restrict__ src_k,
                                   const float4* __restrict__ src_v,
                                   float4* __restrict__ out,
                                   long n4)
{
    __shared__ float4 stage[2][BATCH][BULK_THREADS];   // 32 KB of 320 KB WGP LDS

    const float4* __restrict__ src = (blockIdx.y == 0) ? src_k : src_v;
    float4* __restrict__ dst = out + (size_t)blockIdx.y * (size_t)n4;

    const unsigned tid  = threadIdx.x;
    const size_t g      = (size_t)blockIdx.x * blockDim.x + tid;
    const size_t stride = (size_t)gridDim.x * blockDim.x;
    const size_t step   = stride * BATCH;

    // LDS byte offsets of this thread's slots. Flat shared-aperture addresses
    // carry the LDS byte offset in addr[31:0] (ISA 10.2), so truncation works.
    uint32_t lds_off[2][BATCH];
#pragma unroll
    for (int bf = 0; bf < 2; ++bf)
#pragma unroll
        for (int k = 0; k < BATCH; ++k)
            lds_off[bf][k] = (uint32_t)(uintptr_t)(&stage[bf][k][tid]);

    if (g >= (size_t)n4) return;   // uniform across the wave (n4 % stride == 0)

    // Prologue: fill buffer 0 with the first super-iteration's loads.
#pragma unroll
    for (int k = 0; k < BATCH; ++k) {
        size_t idx = g + (size_t)k * stride;
        if (idx < (size_t)n4) {
            uint64_t ga = (uint64_t)(uintptr_t)(src + idx);
            asm volatile("global_load_async_to_lds_b128 %0, %1, off th:TH_LOAD_NT"
                         :: "v"(lds_off[0][k]), "v"(ga) : "memory");
        }
    }

    int cur = 0;
    for (size_t base = g; base < (size_t)n4; base += step) {
        asm volatile("s_wait_asynccnt 0x0" ::: "memory");  // buf[cur] loaded, prior stores done

        // Stores of buffer `cur` overlap with loads into buffer `cur^1`.
#pragma unroll
        for (int k = 0; k < BATCH; ++k) {
            size_t idx = base + (size_t)k * stride;
            if (idx < (size_t)n4) {
                uint64_t ga = (uint64_t)(uintptr_t)(dst + idx);
                asm volatile("global_store_async_from_lds_b128 %0, %1, off th:TH_STORE_NT"
                             :: "v"(ga), "v"(lds_off[cur][k]) : "memory");
            }
        }

        size_t nbase = base + step;
        if (nbase < (size_t)n4) {
#pragma unroll
            for (int k = 0; k < BATCH; ++k) {
                size_t idx = nbase + (size_t)k * stride;
                if (idx < (size_t)n4) {
                    uint64_t ga = (uint64_t)(uintptr_t)(src + idx);
                    asm volatile("global_load_async_to_lds_b128 %0, %1, off th:TH_LOAD_NT"
                                 :: "v"(lds_off[cur ^ 1][k]), "v"(ga) : "memory");
                }
            }
        }
        cur ^= 1;
    }
    asm volatile("s_wait_asynccnt 0x0" ::: "memory");      // drain final stores
}

// ---------------------------------------------------------------------------
// Kernel 2: overwrite the S_NEW window with new_k/new_v at target_positions[b].
// Tiny (512 KB total) — one float4 per thread. Launched after the bulk copy
// on the same stream -> ordering guaranteed.
// ---------------------------------------------------------------------------
__global__ void kv_scatter_new(const float4* __restrict__ new_k,
                               const float4* __restrict__ new_v,
                               const int* __restrict__ tpos,
                               float4* __restrict__ out)
{
    constexpr int per_t = kL * kB * kH * kSN * kD4;   // 16384
    constexpr int total = 2 * per_t;                  // 32768

    int i = blockIdx.x * blockDim.x + threadIdx.x;
    if (i >= total) return;

    const int t = i / per_t;
    int r = i - t * per_t;                 // flat index into (L,B,H,S_NEW,D4)

    const float4 v = (t == 0) ? new_k[r] : new_v[r];

    const int l  = r / (kB * kH * kSN * kD4);  r -= l * (kB * kH * kSN * kD4);
    const int b  = r / (kH * kSN * kD4);       r -= b * (kH * kSN * kD4);
    const int h  = r / (kSN * kD4);            r -= h * (kSN * kD4);
    const int s  = r / kD4;
    const int dv = r - s * kD4;

    const int off = tpos[b];               // target_positions is (B,1) int32

    size_t o = ((((size_t)t * kL + l) * kB + b) * kH + h);
    o = (o * kSF + (size_t)(off + s)) * kD4 + dv;
    out[o] = v;
}

extern "C" void kernel_launch(void* const* d_in, const int* in_sizes, int n_in,
                              void* d_out, int out_size, void* d_ws, size_t ws_size,
                              hipStream_t stream)
{
    (void)in_sizes; (void)n_in; (void)out_size; (void)d_ws; (void)ws_size;

    const float4* past_k = (const float4*)d_in[0];
    const float4* past_v = (const float4*)d_in[1];
    const float4* new_k  = (const float4*)d_in[2];
    const float4* new_v  = (const float4*)d_in[3];
    const int*    tpos   = (const int*)d_in[4];
    float4*       out    = (float4*)d_out;

    // Bulk copy: 2 x 16,777,216 float4. 4096 blocks x 256 threads -> each
    // thread moves 16 chunks = 4 double-buffered pipeline iterations.
    dim3 grid(BULK_BLOCKS, 2, 1);
    kv_bulk_copy_async<<<grid, BULK_THREADS, 0, stream>>>(past_k, past_v, out, kN4);

    // Scatter of the new rows (32768 float4 writes).
    constexpr int total = 2 * kL * kB * kH * kSN * kD4;
    kv_scatter_new<<<(total + 255) / 256, 256, 0, stream>>>(new_k, new_v, tpos, out);
}